// DynamicRouting_64003602645476
// MI455X (gfx1250) — compile-verified
//
#include <hip/hip_runtime.h>
#include <hip/hip_bf16.h>

typedef float v2f __attribute__((ext_vector_type(2)));
typedef float v8f __attribute__((ext_vector_type(8)));

#if defined(__has_builtin)
#  if __has_builtin(__builtin_amdgcn_wmma_f32_16x16x4_f32)
#    define USE_WMMA_F32 1
#  endif
#endif
#ifndef USE_WMMA_F32
#  define USE_WMMA_F32 0
#endif

#define B_    128
#define J_    10
#define I_    1152
#define N_    16
#define TPB   256
#define WAVES 8
#define SLAB  (I_ / WAVES)   // 144 i-values per wave
#define CHUNK (SLAB / 4)     // 36 WMMAs per wave per j

// One workgroup per batch element b. All 3 routing iterations fused; b_vec and
// softmax coefficients live in LDS (~100 KB << 320 KB/WGP). u_hat streamed from
// global (L2-resident after first pass: 94 MB < 192 MB L2).
__global__ __launch_bounds__(TPB) void DynamicRouting_64003602645476_kernel(
    const float* __restrict__ u_hat,   // [B,J,I,N]
    const float* __restrict__ bias,    // [J,N]
    float* __restrict__ out,           // [B,J,N] then +1 scalar slot
    float* __restrict__ ws)            // [B] per-batch dropped-capsule counts
{
    __shared__ float b_lds[J_ * I_];        // routing logits b_vec for this b
    __shared__ float c_lds[J_ * I_];        // softmax(b_vec) over j
    __shared__ float s_part[J_][WAVES][16]; // per-wave partial s
    __shared__ float s_lds[J_ * N_];
    __shared__ float v_lds[J_ * N_];
    __shared__ float scale_lds[J_];
    __shared__ float red[TPB];
    __shared__ float avgb[J_];
    __shared__ float keep[J_];

    const int b    = blockIdx.x;
    const int tid  = threadIdx.x;
    const int lane = tid & 31;
    const int wave = tid >> 5;
    const float* ub = u_hat + (size_t)b * (J_ * I_ * N_);

    for (int t = tid; t < J_ * I_; t += TPB) b_lds[t] = 0.0f;
    __syncthreads();

    for (int it = 0; it < 3; ++it) {
        // ---- (1) softmax over j for each i ----
        for (int i = tid; i < I_; i += TPB) {
            float mx = b_lds[i];
            for (int j = 1; j < J_; ++j) mx = fmaxf(mx, b_lds[j * I_ + i]);
            float den = 0.0f;
            for (int j = 0; j < J_; ++j) den += __expf(b_lds[j * I_ + i] - mx);
            float r = 1.0f / den;
            for (int j = 0; j < J_; ++j)
                c_lds[j * I_ + i] = __expf(b_lds[j * I_ + i] - mx) * r;
        }
        __syncthreads();

        // ---- (2) s[j,n] = sum_i c[j,i] * u_hat[j,i,n] via f32 WMMA ----
        // A(16x4): c[i0..i0+3] broadcast to all 16 rows.
        //   lanes 0-15: VGPR0=K0, VGPR1=K1; lanes 16-31: VGPR0=K2, VGPR1=K3.
        // B(4x16): u_hat rows i0..i0+3, lane = column n, lane-half = K-half.
        // Every D row equals partial s; row M=0 sits in acc[0], lanes 0-15.
        for (int j = 0; j < J_; ++j) {
            const float* uj = ub + (size_t)j * (I_ * N_);
            const float* cj = c_lds + j * I_;
            const int off = (lane >> 4) * 2;  // 0 or 2
            const int col = lane & 15;
#if USE_WMMA_F32
            v8f acc = {};
            for (int ch = 0; ch < CHUNK; ++ch) {
                const int i0 = wave * SLAB + ch * 4;
                v2f a, bm;
                a[0] = cj[i0 + off];
                a[1] = cj[i0 + off + 1];
                const float* up = uj + (size_t)(i0 + off) * N_ + col;
                bm[0] = up[0];     // row i0+off   , col n  (coalesced 128B/wave)
                bm[1] = up[N_];    // row i0+off+1 , col n
                acc = __builtin_amdgcn_wmma_f32_16x16x4_f32(
                    false, a, false, bm, (short)0, acc, false, false);
            }
            if (lane < 16) s_part[j][wave][lane] = acc[0];
#else
            // VALU fallback: half-wave splits the slab, shuffle-combine.
            const int n    = lane & 15;
            const int ibeg = wave * SLAB + (lane >> 4) * (SLAB / 2);
            float sacc = 0.0f;
            for (int i = ibeg; i < ibeg + SLAB / 2; ++i)
                sacc += cj[i] * uj[(size_t)i * N_ + n];
            sacc += __shfl_xor(sacc, 16, 32);
            if (lane < 16) s_part[j][wave][n] = sacc;
#endif
        }
        __syncthreads();

        // ---- (3) reduce partials, add bias, squash ----
        if (tid < J_ * N_) {
            const int j = tid >> 4, n = tid & 15;
            float s = 0.0f;
            for (int w = 0; w < WAVES; ++w) s += s_part[j][w][n];
            s_lds[tid] = s + bias[tid];
        }
        __syncthreads();
        if (tid < J_) {
            float sq = 0.0f;
            for (int n = 0; n < N_; ++n) {
                const float x = s_lds[tid * N_ + n];
                sq += x * x;
            }
            scale_lds[tid] = (sq / (1.0f + sq)) * rsqrtf(sq + 1e-9f);
        }
        __syncthreads();
        if (tid < J_ * N_) {
            const float v = scale_lds[tid >> 4] * s_lds[tid];
            v_lds[tid] = v;
            if (it == 2) out[(size_t)b * (J_ * N_) + tid] = v;
        }
        __syncthreads();

        if (it < 2) {
            // ---- (4) agreement: b_vec[j,i] += u_hat[j,i,:] . v[j,:] ----
            for (int t = tid; t < J_ * I_; t += TPB) {
                const int j = t / I_, i = t % I_;
                const float4* up = (const float4*)(ub + ((size_t)j * I_ + i) * N_);
                const float4* vp = (const float4*)(v_lds + j * N_);
                const float4 u0 = up[0], u1 = up[1], u2 = up[2], u3 = up[3];
                const float4 w0 = vp[0], w1 = vp[1], w2 = vp[2], w3 = vp[3];
                float d = u0.x * w0.x + u0.y * w0.y + u0.z * w0.z + u0.w * w0.w;
                d += u1.x * w1.x + u1.y * w1.y + u1.z * w1.z + u1.w * w1.w;
                d += u2.x * w2.x + u2.y * w2.y + u2.z * w2.z + u2.w * w2.w;
                d += u3.x * w3.x + u3.y * w3.y + u3.z * w3.z + u3.w * w3.w;
                b_lds[t] += d;
            }
            __syncthreads();

            if (it == 0) {
                // ---- (5) logsumexp threshold mask on avg_b over i ----
                for (int j = 0; j < J_; ++j) {
                    float p = 0.0f;
                    for (int i = tid; i < I_; i += TPB) p += b_lds[j * I_ + i];
                    red[tid] = p;
                    __syncthreads();
                    for (int s = TPB / 2; s > 0; s >>= 1) {
                        if (tid < s) red[tid] += red[tid + s];
                        __syncthreads();
                    }
                    if (tid == 0) avgb[j] = red[0] * (1.0f / (float)I_);
                    __syncthreads();
                }
                if (tid == 0) {
                    float m = avgb[0];
                    for (int j = 1; j < J_; ++j) m = fmaxf(m, avgb[j]);
                    float se = 0.0f;
                    for (int j = 0; j < J_; ++j) se += __expf(avgb[j] - m);
                    const float thr = __logf(0.0999f) + (m + __logf(se));
                    float cnt = 0.0f;
                    for (int j = 0; j < J_; ++j) {
                        const float k = (avgb[j] > thr) ? 1.0f : 0.0f;
                        keep[j] = k;
                        cnt += 1.0f - k;
                    }
                    ws[b] = cnt;  // deterministic per-b count, reduced later
                }
                __syncthreads();
                for (int t = tid; t < J_ * I_; t += TPB)
                    b_lds[t] *= keep[t / I_];
                __syncthreads();
            }
        }
    }
}

// Deterministic single-thread reduction of the 128 per-batch dropped counts.
__global__ void DynamicRouting_finalize_kernel(const float* __restrict__ ws,
                                               float* __restrict__ out)
{
    if (threadIdx.x == 0 && blockIdx.x == 0) {
        float tot = 0.0f;
        for (int b = 0; b < B_; ++b) tot += ws[b];
        out[B_ * J_ * N_] = tot * (1.0f / (float)(B_ * J_));
    }
}

extern "C" void kernel_launch(void* const* d_in, const int* in_sizes, int n_in,
                              void* d_out, int out_size, void* d_ws, size_t ws_size,
                              hipStream_t stream) {
    const float* u_hat = (const float*)d_in[0];  // [128,10,1152,16] f32
    const float* bias  = (const float*)d_in[1];  // [10,16] f32
    // d_in[2] = iters (==3, matching the reference; fixed at compile time)
    float* out = (float*)d_out;
    float* ws  = (float*)d_ws;

    DynamicRouting_64003602645476_kernel<<<B_, TPB, 0, stream>>>(u_hat, bias, out, ws);
    DynamicRouting_finalize_kernel<<<1, 32, 0, stream>>>(ws, out);
}